// Transformer_Block_11098195493453
// MI455X (gfx1250) — compile-verified
//
#include <hip/hip_runtime.h>
#include <cstdint>

#define BB 4
#define CC 64
#define HH 256
#define WW 256
#define NN 65536      // HH*WW
#define NHEADS 16
#define NHID 16
#define NEXP 3

typedef float v2f __attribute__((ext_vector_type(2)));
typedef float v8f __attribute__((ext_vector_type(8)));

__device__ inline v8f wmma4(v2f a, v2f b, v8f c) {
  // D = A(16x4 f32) * B(4x16 f32) + C(16x16 f32)
  return __builtin_amdgcn_wmma_f32_16x16x4_f32(false, a, false, b, (short)0, c, false, false);
}
__device__ inline v8f v8zero() { v8f z = {0.f,0.f,0.f,0.f,0.f,0.f,0.f,0.f}; return z; }
__device__ inline v2f ld2(const float* p) { v2f t; t.x = p[0]; t.y = p[1]; return t; }
__device__ inline float gelu_exact(float x) { return 0.5f*x*(1.0f + erff(x*0.7071067811865475f)); }

// ---------------------------------------------------------------------------
// K0: zero small accumulators
__global__ void k_zero(float* __restrict__ p, int n) {
  int i = blockIdx.x*blockDim.x + threadIdx.x;
  if (i < n) p[i] = 0.f;
}

// ---------------------------------------------------------------------------
// K2: per-pixel LN1 (registers+LDS only), xg1 channel sums, and per-head Gram
// matrices Sqk / diag(Sqq) / diag(Skk) for all 3 experts via f32 WMMA.
// Block: 128 threads (4 waves), 128 pixels. Wave w covers channels 16w..16w+15
// (heads 4w..4w+3) for both Q and K.
__global__ __launch_bounds__(128)
void k_gram_xg1(const float* __restrict__ x,
                const float* __restrict__ ln1w, const float* __restrict__ ln1b,
                const float* __restrict__ qkv,
                float* __restrict__ gram, float* __restrict__ xg1sum)
{
  __shared__ float y[128*66];     // [pixel][channel], padded stride 66
  __shared__ float lw[CC], lb[CC];
  const int tid = threadIdx.x;
  const int b  = blockIdx.y;
  const int n0 = blockIdx.x * 128;
  if (tid < CC) { lw[tid] = ln1w[tid]; lb[tid] = ln1b[tid]; }
  __syncthreads();
  { // load + LayerNorm, one pixel per thread
    const int p = tid;
    const float* xp = x + ((size_t)b*CC)*NN + n0 + p;
    float s = 0.f, s2 = 0.f;
    for (int c = 0; c < CC; ++c) {
      float v = xp[(size_t)c*NN];
      y[p*66+c] = v; s += v; s2 += v*v;
    }
    float mu = s*(1.f/CC);
    float rs = rsqrtf(s2*(1.f/CC) - mu*mu + 1e-5f);
    for (int c = 0; c < CC; ++c)
      y[p*66+c] = (y[p*66+c]-mu)*rs*lw[c] + lb[c];
  }
  __syncthreads();
  if (tid < CC) { // xg1 partial channel sums
    float s = 0.f;
    for (int p = 0; p < 128; ++p) s += y[p*66+tid];
    atomicAdd(&xg1sum[b*CC+tid], s);
  }
  const int wv = tid>>5, lane = tid&31, lm = lane&15, lh = lane>>4;
  const int m  = 16*wv + lm;    // output channel row for A operand
  for (int e = 0; e < NEXP; ++e) {
    const float* W = qkv + (size_t)e*(3*CC)*CC;
    v2f aQ[16], aK[16];
    #pragma unroll
    for (int kk = 0; kk < 16; ++kk) {
      aQ[kk] = ld2(W + (size_t)m*CC      + 4*kk + 2*lh);
      aK[kk] = ld2(W + (size_t)(CC+m)*CC + 4*kk + 2*lh);
    }
    float aqk[32], aqq[8], akk[8];
    #pragma unroll
    for (int i=0;i<32;++i) aqk[i]=0.f;
    #pragma unroll
    for (int i=0;i<8;++i){ aqq[i]=0.f; akk[i]=0.f; }
    for (int s = 0; s < 8; ++s) {       // 8 subtiles of 16 pixels
      v2f bb[16];
      #pragma unroll
      for (int kk=0;kk<16;++kk) bb[kk] = ld2(&y[(s*16+lm)*66 + 4*kk + 2*lh]);
      v8f q = v8zero(), k = v8zero();
      #pragma unroll
      for (int kk=0;kk<16;++kk) q = wmma4(aQ[kk], bb[kk], q);
      #pragma unroll
      for (int kk=0;kk<16;++kk) k = wmma4(aK[kk], bb[kk], k);
      // per lane: VGPRs 0..3 = head A channels, 4..7 = head B channels
      #pragma unroll
      for (int h2=0; h2<2; ++h2)
        #pragma unroll
        for (int c=0;c<4;++c) {
          float qc = q[h2*4+c], kc = k[h2*4+c];
          aqq[h2*4+c] += qc*qc;
          akk[h2*4+c] += kc*kc;
          #pragma unroll
          for (int d=0;d<4;++d) aqk[h2*16+c*4+d] += qc * k[h2*4+d];
        }
    }
    #pragma unroll
    for (int msk=8; msk>=1; msk>>=1) {  // reduce over 16 lanes of each half
      #pragma unroll
      for (int i=0;i<32;++i) aqk[i] += __shfl_xor(aqk[i], msk, 32);
      #pragma unroll
      for (int i=0;i<8;++i){ aqq[i] += __shfl_xor(aqq[i], msk, 32);
                             akk[i] += __shfl_xor(akk[i], msk, 32); }
    }
    if (lm == 0) {
      #pragma unroll
      for (int h2=0; h2<2; ++h2) {
        int head = 4*wv + 2*lh + h2;
        float* g = gram + (((size_t)b*NEXP + e)*NHEADS + head)*24;
        for (int i=0;i<16;++i) atomicAdd(&g[i], aqk[h2*16+i]);
        for (int i=0;i<4;++i){ atomicAdd(&g[16+i], aqq[h2*4+i]);
                               atomicAdd(&g[20+i], akk[h2*4+i]); }
      }
    }
  }
}

// ---------------------------------------------------------------------------
// K1/K6: routing MLP (per batch): gelu(W1*[xg,hidden]+b1) -> relu(W3*h+b3)
__global__ void k_routing(const float* __restrict__ xgsum, const float* __restrict__ hid_in,
                          const float* __restrict__ w1, const float* __restrict__ b1,
                          const float* __restrict__ w3, const float* __restrict__ b3,
                          float* __restrict__ hid_out, float* __restrict__ logit)
{
  __shared__ float inv[CC+NHID];
  __shared__ float h1s[NHID];
  const int b = blockIdx.x, t = threadIdx.x;
  for (int i=t;i<CC;i+=32) inv[i] = xgsum[b*CC+i]*(1.f/(float)NN);
  if (t < NHID) inv[CC+t] = hid_in[b*NHID+t];
  __syncthreads();
  if (t < NHID) {
    float s = b1[t];
    for (int i=0;i<CC+NHID;++i) s += w1[t*(CC+NHID)+i]*inv[i];
    s = gelu_exact(s);
    h1s[t] = s; hid_out[b*NHID+t] = s;
  }
  __syncthreads();
  if (t < NEXP) {
    float s = b3[t];
    for (int j=0;j<NHID;++j) s += w3[t*NHID+j]*h1s[j];
    logit[b*NEXP+t] = fmaxf(s, 0.f);
  }
}

// ---------------------------------------------------------------------------
// K3: 4x4 softmax per (b,e,head) from Gram stats (with l2-normalize semantics)
__global__ void k_attn_softmax(const float* __restrict__ gram,
                               const float* __restrict__ temp,
                               float* __restrict__ attn)
{
  int idx = threadIdx.x;
  if (idx >= BB*NEXP*NHEADS) return;
  int h = idx % NHEADS;
  int e = (idx / NHEADS) % NEXP;
  const float* g = gram + (size_t)idx*24;
  float tp = temp[e*NHEADS + h];
  float qn[4], kn[4];
  for (int c=0;c<4;++c){ qn[c]=fmaxf(sqrtf(g[16+c]),1e-12f);
                         kn[c]=fmaxf(sqrtf(g[20+c]),1e-12f); }
  float* a = attn + (size_t)idx*16;
  for (int c=0;c<4;++c){
    float v[4]; float mx = -3.4e38f;
    for (int d=0;d<4;++d){ v[d]=tp*g[c*4+d]/(qn[c]*kn[d]); mx=fmaxf(mx,v[d]); }
    float ss=0.f;
    for (int d=0;d<4;++d){ v[d]=expf(v[d]-mx); ss+=v[d]; }
    float is = 1.f/ss;
    for (int d=0;d<4;++d) a[c*4+d]=v[d]*is;
  }
}

// ---------------------------------------------------------------------------
// K3b: Meff[b,e] = blockdiag(attn) @ Wv  (64x64 per (b,e))
__global__ void k_meff(const float* __restrict__ attn, const float* __restrict__ qkv,
                       float* __restrict__ meff)
{
  int be = blockIdx.x;              // b*NEXP + e
  int e  = be % NEXP;
  const float* a  = attn + (size_t)be*NHEADS*16;
  const float* Wv = qkv + (size_t)e*3*CC*CC + (size_t)2*CC*CC;
  float* M = meff + (size_t)be*CC*CC;
  for (int idx = threadIdx.x; idx < CC*CC; idx += blockDim.x) {
    int row = idx >> 6, col = idx & 63;
    int h = row >> 2, c = row & 3;
    float s = 0.f;
    for (int d=0;d<4;++d) s += a[h*16 + c*4 + d] * Wv[(size_t)(h*4+d)*CC + col];
    M[idx] = s;
  }
}

// ---------------------------------------------------------------------------
// K4: fused attention bank. 4x8 center tile, 6x10 halo. Per expert:
//   t = Meff@LN1(x) (WMMA) -> 3x3 depthwise in LDS -> Wproj@u (WMMA)
//   acc += logit1 * result.  Epilogue: out1 = acc + beta*x, plus LN2 partial
//   channel sums for xg2.
__global__ __launch_bounds__(128)
void k_attn_fused(const float* __restrict__ x,
                  const float* __restrict__ ln1w, const float* __restrict__ ln1b,
                  const float* __restrict__ ln2w, const float* __restrict__ ln2b,
                  const float* __restrict__ beta,
                  const float* __restrict__ dwc, const float* __restrict__ proj,
                  const float* __restrict__ meff, const float* __restrict__ logit1,
                  float* __restrict__ out1, float* __restrict__ xg2sum)
{
  __shared__ float y[64*66];     // LN1(x) halo, [p][c] (60 used + pad)
  __shared__ float tb[CC*64];    // t, [c][halo p] (60 used + pad)
  __shared__ float ub[32*66];    // conv output / out1 tile, [pc][c]
  __shared__ float wd[CC*9];
  __shared__ float lw[CC], lb[CC];
  __shared__ float muS[32], rsS[32];
  const int tid = threadIdx.x, wv = tid>>5, lane = tid&31, lm = lane&15, lh = lane>>4;
  const int b = blockIdx.z, ty = blockIdx.y, tx = blockIdx.x;
  if (tid < CC) { lw[tid]=ln1w[tid]; lb[tid]=ln1b[tid]; }
  __syncthreads();
  if (tid < 64) { // halo load + LN1 (zeros outside image & in pad slots)
    const int p = tid;
    bool valid = false; int n = 0;
    if (p < 60) {
      int hy = p/10, hx = p%10;
      int gy = ty*4 - 1 + hy, gx = tx*8 - 1 + hx;
      if (gy>=0 && gy<HH && gx>=0 && gx<WW) { valid = true; n = gy*WW + gx; }
    }
    if (valid) {
      const float* xp = x + ((size_t)b*CC)*NN + n;
      float s=0.f, s2=0.f;
      for (int c=0;c<CC;++c){ float v=xp[(size_t)c*NN]; y[p*66+c]=v; s+=v; s2+=v*v; }
      float mu = s*(1.f/CC), rs = rsqrtf(s2*(1.f/CC)-mu*mu+1e-5f);
      for (int c=0;c<CC;++c) y[p*66+c] = (y[p*66+c]-mu)*rs*lw[c] + lb[c];
    } else {
      for (int c=0;c<CC;++c) y[p*66+c] = 0.f;
    }
  }
  __syncthreads();
  v8f acc[2]; acc[0]=v8zero(); acc[1]=v8zero();
  const int m = 16*wv + lm;
  for (int e=0;e<NEXP;++e) {
    const float* Mrow = meff + ((size_t)(b*NEXP+e)*CC + m)*CC;
    v2f aM[16];
    #pragma unroll
    for (int kk=0;kk<16;++kk) aM[kk] = ld2(Mrow + 4*kk + 2*lh);
    #pragma unroll
    for (int s=0;s<4;++s) {        // t over 64 (padded) halo pixels
      v8f d = v8zero();
      #pragma unroll
      for (int kk=0;kk<16;++kk) {
        v2f bv = ld2(&y[(s*16+lm)*66 + 4*kk + 2*lh]);
        d = wmma4(aM[kk], bv, d);
      }
      int p = s*16 + lm;
      #pragma unroll
      for (int r=0;r<8;++r) tb[(16*wv + 8*lh + r)*64 + p] = d[r];
    }
    for (int i=tid;i<CC*9;i+=128) wd[i] = dwc[e*CC*9 + i];
    __syncthreads();
    for (int idx=tid; idx<2048; idx+=128) {   // depthwise 3x3, SAME (halo pre-zeroed)
      int c = idx & 63, pc = idx >> 6;
      int cy = pc >> 3, cx = pc & 7;
      const float* w9 = &wd[c*9];
      const float* tc = &tb[c*64];
      float u = 0.f;
      #pragma unroll
      for (int dy=0;dy<3;++dy)
        #pragma unroll
        for (int dx=0;dx<3;++dx)
          u += tc[(cy+dy)*10 + cx+dx] * w9[dy*3+dx];
      ub[pc*66 + c] = u;
    }
    __syncthreads();
    const float* Prow = proj + ((size_t)e*CC + m)*CC;
    v2f aP[16];
    #pragma unroll
    for (int kk=0;kk<16;++kk) aP[kk] = ld2(Prow + 4*kk + 2*lh);
    const float lg = logit1[b*NEXP + e];
    #pragma unroll
    for (int s=0;s<2;++s) {        // proj over 32 center pixels
      v8f d = v8zero();
      #pragma unroll
      for (int kk=0;kk<16;++kk) {
        v2f bv = ld2(&ub[(s*16+lm)*66 + 4*kk + 2*lh]);
        d = wmma4(aP[kk], bv, d);
      }
      #pragma unroll
      for (int r=0;r<8;++r) acc[s][r] += lg * d[r];
    }
  }
  __syncthreads();
  #pragma unroll
  for (int s=0;s<2;++s) {          // epilogue: out1 = acc + beta*skip
    int pc = s*16 + lm;
    int cy = pc>>3, cx = pc&7;
    int n = (ty*4+cy)*WW + tx*8+cx;
    #pragma unroll
    for (int r=0;r<8;++r) {
      int c = 16*wv + 8*lh + r;
      float v = acc[s][r] + beta[c]*x[((size_t)b*CC + c)*NN + n];
      out1[((size_t)b*CC + c)*NN + n] = v;
      ub[pc*66 + c] = v;
    }
  }
  __syncthreads();
  if (tid < 32) {                  // per-pixel LN2 stats for xg2
    float s=0.f, s2=0.f;
    for (int c=0;c<CC;++c){ float v=ub[tid*66+c]; s+=v; s2+=v*v; }
    float mu = s*(1.f/CC);
    muS[tid]=mu; rsS[tid]=rsqrtf(s2*(1.f/CC)-mu*mu+1e-5f);
  }
  __syncthreads();
  if (tid < CC) {
    float s=0.f;
    for (int p=0;p<32;++p) s += (ub[p*66+tid]-muS[p])*rsS[p];
    atomicAdd(&xg2sum[b*CC+tid], ln2w[tid]*s + 32.f*ln2b[tid]);
  }
}

// ---------------------------------------------------------------------------
// K7: fused MLP bank. 32-pixel tile: LN2 in LDS, per-expert
// h=gelu(W1 y + b1) into LDS, z = W2 h + b2 via WMMA, logit2-weighted acc,
// final out = acc + gamma*skip written straight to d_out.
__global__ __launch_bounds__(128)
void k_mlp_fused(const float* __restrict__ out1,
                 const float* __restrict__ ln2w, const float* __restrict__ ln2b,
                 const float* __restrict__ gamma, const float* __restrict__ logit2,
                 const float* __restrict__ w1_0, const float* __restrict__ b1_0,
                 const float* __restrict__ w2_0, const float* __restrict__ b2_0,
                 const float* __restrict__ w1_1, const float* __restrict__ b1_1,
                 const float* __restrict__ w2_1, const float* __restrict__ b2_1,
                 const float* __restrict__ w1_2, const float* __restrict__ b1_2,
                 const float* __restrict__ w2_2, const float* __restrict__ b2_2,
                 float* __restrict__ out)
{
  __shared__ float y2[32*66];
  __shared__ float hb[32*258];     // up to hf=256 hidden channels, padded
  const int tid=threadIdx.x, wv=tid>>5, lane=tid&31, lm=lane&15, lh=lane>>4;
  const int b = blockIdx.y; const int n0 = blockIdx.x*32;
  if (tid < 32) {
    const float* xp = out1 + ((size_t)b*CC)*NN + n0 + tid;
    float s=0.f, s2=0.f;
    for (int c=0;c<CC;++c){ float v=xp[(size_t)c*NN]; y2[tid*66+c]=v; s+=v; s2+=v*v; }
    float mu = s*(1.f/CC), rs = rsqrtf(s2*(1.f/CC)-mu*mu+1e-5f);
    for (int c=0;c<CC;++c) y2[tid*66+c] = (y2[tid*66+c]-mu)*rs*ln2w[c] + ln2b[c];
  }
  __syncthreads();
  const float* W1[NEXP] = {w1_0, w1_1, w1_2};
  const float* B1[NEXP] = {b1_0, b1_1, b1_2};
  const float* W2[NEXP] = {w2_0, w2_1, w2_2};
  const float* B2[NEXP] = {b2_0, b2_1, b2_2};
  v8f acc[2]; acc[0]=v8zero(); acc[1]=v8zero();
  for (int e=0;e<NEXP;++e) {
    const int hf = CC << e;        // 64,128,256
    for (int cc=0; cc < (hf>>6); ++cc) {
      const int hc0 = cc*64 + 16*wv;
      const float* Arow = W1[e] + (size_t)(hc0+lm)*CC;
      v2f aH[16];
      #pragma unroll
      for (int kk=0;kk<16;++kk) aH[kk] = ld2(Arow + 4*kk + 2*lh);
      #pragma unroll
      for (int s=0;s<2;++s) {
        v8f d = v8zero();
        #pragma unroll
        for (int kk=0;kk<16;++kk) {
          v2f bv = ld2(&y2[(s*16+lm)*66 + 4*kk + 2*lh]);
          d = wmma4(aH[kk], bv, d);
        }
        int p = s*16+lm;
        #pragma unroll
        for (int r=0;r<8;++r) {
          int hc = hc0 + 8*lh + r;
          hb[p*258 + hc] = gelu_exact(d[r] + B1[e][hc]);
        }
      }
    }
    __syncthreads();
    const float lg = logit2[b*NEXP+e];
    const int nk = hf >> 2;
    #pragma unroll
    for (int s=0;s<2;++s) {
      v8f d = v8zero();
      const float* Arow = W2[e] + (size_t)(16*wv+lm)*hf;
      for (int kk=0;kk<nk;++kk) {
        int col = 4*kk + 2*lh;
        v2f av = ld2(Arow + col);
        v2f bv = ld2(&hb[(s*16+lm)*258 + col]);
        d = wmma4(av, bv, d);
      }
      #pragma unroll
      for (int r=0;r<8;++r) {
        int c = 16*wv + 8*lh + r;
        acc[s][r] += lg * (d[r] + B2[e][c]);
      }
    }
    __syncthreads();
  }
  #pragma unroll
  for (int s=0;s<2;++s) {
    int n = n0 + s*16 + lm;
    #pragma unroll
    for (int r=0;r<8;++r) {
      int c = 16*wv + 8*lh + r;
      size_t off = ((size_t)b*CC + c)*NN + n;
      out[off] = acc[s][r] + gamma[c]*out1[off];
    }
  }
}

// ---------------------------------------------------------------------------
// K8: pack hidden + logits into d_out tail
__global__ void k_pack(const float* __restrict__ hid2,
                       const float* __restrict__ lg1, const float* __restrict__ lg2,
                       float* __restrict__ out)
{
  int t = threadIdx.x;
  float* o = out + (size_t)BB*CC*NN;
  if (t < BB*NHID) o[t] = hid2[t];
  if (t < BB*NEXP) { o[BB*NHID + 2*t] = lg1[t]; o[BB*NHID + 2*t + 1] = lg2[t]; }
}

// ---------------------------------------------------------------------------
extern "C" void kernel_launch(void* const* d_in, const int* in_sizes, int n_in,
                              void* d_out, int out_size, void* d_ws, size_t ws_size,
                              hipStream_t stream)
{
  (void)in_sizes; (void)n_in; (void)out_size; (void)ws_size;
  const float* x      = (const float*)d_in[0];
  const float* hid_in = (const float*)d_in[1];
  const float* ln1w = (const float*)d_in[2];
  const float* ln1b = (const float*)d_in[3];
  const float* ln2w = (const float*)d_in[4];
  const float* ln2b = (const float*)d_in[5];
  const float* beta = (const float*)d_in[6];
  const float* gmma = (const float*)d_in[7];
  const float* temp = (const float*)d_in[8];
  const float* qkv  = (const float*)d_in[9];
  const float* dwc  = (const float*)d_in[10];
  const float* proj = (const float*)d_in[11];
  const float* ar1w = (const float*)d_in[12];
  const float* ar1b = (const float*)d_in[13];
  const float* ar3w = (const float*)d_in[14];
  const float* ar3b = (const float*)d_in[15];
  const float* mr1w = (const float*)d_in[16];
  const float* mr1b = (const float*)d_in[17];
  const float* mr3w = (const float*)d_in[18];
  const float* mr3b = (const float*)d_in[19];
  const float* w1_0 = (const float*)d_in[20];
  const float* b1_0 = (const float*)d_in[21];
  const float* w2_0 = (const float*)d_in[22];
  const float* b2_0 = (const float*)d_in[23];
  const float* w1_1 = (const float*)d_in[24];
  const float* b1_1 = (const float*)d_in[25];
  const float* w2_1 = (const float*)d_in[26];
  const float* b2_1 = (const float*)d_in[27];
  const float* w1_2 = (const float*)d_in[28];
  const float* b1_2 = (const float*)d_in[29];
  const float* w2_2 = (const float*)d_in[30];
  const float* b2_2 = (const float*)d_in[31];

  float* out = (float*)d_out;
  float* ws  = (float*)d_ws;
  float* out1 = ws;                                   // B*C*N
  float* meff = out1 + (size_t)BB*CC*NN;              // B*NEXP*64*64
  float* gram = meff + (size_t)BB*NEXP*CC*CC;         // B*NEXP*16*24
  float* xg1  = gram + (size_t)BB*NEXP*NHEADS*24;     // B*C
  float* xg2  = xg1 + BB*CC;                          // B*C
  float* attw = xg2 + BB*CC;                          // B*NEXP*16*16
  float* h1   = attw + (size_t)BB*NEXP*NHEADS*16;     // B*HID
  float* h2   = h1 + BB*NHID;                         // B*HID
  float* l1   = h2 + BB*NHID;                         // B*NEXP
  float* l2   = l1 + BB*NEXP;                         // B*NEXP

  const int nz = BB*NEXP*NHEADS*24 + 2*BB*CC;         // gram + xg1 + xg2 (contiguous)
  k_zero<<<(nz+255)/256, 256, 0, stream>>>(gram, nz);

  k_gram_xg1<<<dim3(NN/128, BB), 128, 0, stream>>>(x, ln1w, ln1b, qkv, gram, xg1);
  k_routing<<<BB, 32, 0, stream>>>(xg1, hid_in, ar1w, ar1b, ar3w, ar3b, h1, l1);
  k_attn_softmax<<<1, 256, 0, stream>>>(gram, temp, attw);
  k_meff<<<BB*NEXP, 256, 0, stream>>>(attw, qkv, meff);
  k_attn_fused<<<dim3(WW/8, HH/4, BB), 128, 0, stream>>>(
      x, ln1w, ln1b, ln2w, ln2b, beta, dwc, proj, meff, l1, out1, xg2);
  k_routing<<<BB, 32, 0, stream>>>(xg2, h1, mr1w, mr1b, mr3w, mr3b, h2, l2);
  k_mlp_fused<<<dim3(NN/32, BB), 128, 0, stream>>>(
      out1, ln2w, ln2b, gmma, l2,
      w1_0, b1_0, w2_0, b2_0,
      w1_1, b1_1, w2_1, b2_1,
      w1_2, b1_2, w2_2, b2_2, out);
  k_pack<<<1, 64, 0, stream>>>(h2, l1, l2, out);
}